// TripletLossPRCC_38749194945227
// MI455X (gfx1250) — compile-verified
//
#include <hip/hip_runtime.h>
#include <hip/hip_bf16.h>

typedef __attribute__((ext_vector_type(2))) float v2f;
typedef __attribute__((ext_vector_type(8))) float v8f;

#define MARGIN_F 0.3f
#define LDS_PAD 4       // A-panel row stride = d + 4 words -> conflict-free frags
#define MROWS   64      // rows per block (A panel in LDS = 64*(512+4)*4 = 132 KB)

// ---------------------------------------------------------------------------
// Kernel 0: sq[i] = sum_k X[i][k]^2  (one wave32 per row); zeroes num/cnt.
// ---------------------------------------------------------------------------
__global__ void __launch_bounds__(256) sq_kernel(const float* __restrict__ X,
                                                 float* __restrict__ ws,
                                                 int n, int d) {
  if (blockIdx.x == 0 && threadIdx.x == 0) { ws[0] = 0.0f; ws[1] = 0.0f; }
  float* sq = ws + 2;
  const int lane = threadIdx.x & 31;
  const int wave = threadIdx.x >> 5;
  const int row  = blockIdx.x * 8 + wave;
  if (row >= n) return;
  const float* xr = X + (size_t)row * d;
  float s = 0.0f;
  for (int c = lane * 4; c < d; c += 128) {
    float4 v = *(const float4*)(xr + c);
    s += v.x * v.x + v.y * v.y + v.z * v.z + v.w * v.w;
  }
#pragma unroll
  for (int off = 16; off >= 1; off >>= 1) s += __shfl_xor(s, off, 32);
  if (lane == 0) sq[row] = s;
}

// ---------------------------------------------------------------------------
// Double-buffered GEMM helpers: one wave computes a 16x64 strip of the Gram
// matrix with v_wmma_f32_16x16x4_f32. B fragments for a 16-deep K chunk are
// prefetched into a register ping/pong pair so load latency overlaps WMMA.
// ---------------------------------------------------------------------------
__device__ __forceinline__ void load_b_chunk(v2f (&buf)[4][4],
                                             const float* const (&bp)[4], int kk) {
#pragma unroll
  for (int t = 0; t < 4; ++t)
#pragma unroll
    for (int ks = 0; ks < 4; ++ks)
      buf[t][ks] = *(const v2f*)(bp[t] + kk + ks * 4);
}

__device__ __forceinline__ void compute_chunk(v8f (&acc)[4], const v2f (&buf)[4][4],
                                              const float* a_base, int kk) {
#pragma unroll
  for (int ks = 0; ks < 4; ++ks) {
    v2f a = *(const v2f*)(a_base + kk + ks * 4);   // ds_load, immediate offsets
#pragma unroll
    for (int t = 0; t < 4; ++t)
      acc[t] = __builtin_amdgcn_wmma_f32_16x16x4_f32(
          false, a, false, buf[t][ks], (short)0, acc[t], false, false);
  }
}

__device__ __forceinline__ void gemm_group(v8f (&acc)[4], const float* __restrict__ X,
                                           int colBase, int m, int koff,
                                           const float* a_base, int d) {
  const float* bp[4];
#pragma unroll
  for (int t = 0; t < 4; ++t)
    bp[t] = X + (size_t)(colBase + t * 16 + m) * d + koff;

  v2f bA[4][4], bB[4][4];
  load_b_chunk(bA, bp, 0);
  for (int kk = 0; kk < d; kk += 32) {
    load_b_chunk(bB, bp, kk + 16);
    compute_chunk(acc, bA, a_base, kk);
    if (kk + 32 < d) load_b_chunk(bA, bp, kk + 32);
    compute_chunk(acc, bB, a_base, kk + 16);
  }
}

// ---------------------------------------------------------------------------
// Main fused kernel: block = 256 threads (8 wave32) owns a 64-row tile whose
// A panel is staged in dynamic LDS. Wave w: row slice (w>>1)*16, column half
// (w&1). Pass 1: hardest-negative min per row. Pass 2: selected-pair loss.
// ---------------------------------------------------------------------------
__global__ void __launch_bounds__(256) triplet_main(const float* __restrict__ X,
                                                    const int* __restrict__ tgt,
                                                    const int* __restrict__ cam,
                                                    float* __restrict__ ws,
                                                    int n, int d) {
  extern __shared__ float Apanel[];        // MROWS * (d + LDS_PAD) floats
  __shared__ float ldsmin[8][16];
  __shared__ float negmin_s[MROWS];
  __shared__ float psum[8];
  __shared__ float pcnt[8];
  __shared__ int   tgtrow_s[MROWS];
  __shared__ int   camrow_s[MROWS];

  const float* sq = ws + 2;
  const int tid     = threadIdx.x;
  const int lane    = tid & 31;
  const int wave    = tid >> 5;
  const int rowBase = blockIdx.x * MROWS;
  const int lstride = d + LDS_PAD;

  // Stage A panel (MROWS x d) into LDS, float4 per thread.
  const int vecsPerRow = d >> 2;
  for (int i = tid; i < MROWS * vecsPerRow; i += 256) {
    int r = i / vecsPerRow;
    int c = (i - r * vecsPerRow) * 4;
    float4 v = *(const float4*)(X + (size_t)(rowBase + r) * d + c);
    *(float4*)(&Apanel[r * lstride + c]) = v;
  }
  if (tid < MROWS) {
    tgtrow_s[tid] = tgt[rowBase + tid];
    camrow_s[tid] = cam[rowBase + tid];
  }
  __syncthreads();

  // WMMA fragment geometry (wave32, 16x16x4 f32):
  //   A/B: lanes 0-15 hold row/col m=lane with K=0,1; lanes 16-31 K=2,3.
  //   C/D: row = v + 8*(lane>=16), col = lane&15.
  const int m      = lane & 15;
  const int hi     = lane >> 4;
  const int koff   = hi * 2;
  const int rslice = wave >> 1;             // 0..3 -> 16-row slice
  const int colset = wave & 1;              // 0..1 -> column half
  const int rloc0  = rslice * 16;

  float sqrow[8]; int trow[8]; int crow[8];
#pragma unroll
  for (int v = 0; v < 8; ++v) {
    int rl = rloc0 + v + hi * 8;
    sqrow[v] = sq[rowBase + rl];
    trow[v]  = tgtrow_s[rl];
    crow[v]  = camrow_s[rl];
  }

  const int numGroups = n >> 6;             // groups of 64 columns
  const v8f vzero = {0.f, 0.f, 0.f, 0.f, 0.f, 0.f, 0.f, 0.f};
  const float* a_base = &Apanel[(rloc0 + m) * lstride + koff];

  // -------------------- PASS 1: hardest negative per row --------------------
  float minv[8];
#pragma unroll
  for (int v = 0; v < 8; ++v) minv[v] = 3.0e38f;

  for (int g = colset; g < numGroups; g += 2) {
    const int colBase = g * 64;
    v8f acc[4];
#pragma unroll
    for (int t = 0; t < 4; ++t) acc[t] = vzero;
    gemm_group(acc, X, colBase, m, koff, a_base, d);
#pragma unroll
    for (int t = 0; t < 4; ++t) {
      int col   = colBase + t * 16 + m;
      float sqc = sq[col];
      int   tc  = tgt[col];
#pragma unroll
      for (int v = 0; v < 8; ++v) {
        float d2 = sqrow[v] + sqc - 2.0f * acc[t][v];
        float dd = __builtin_sqrtf(fmaxf(d2, 1e-12f));
        if (tc != trow[v]) minv[v] = fminf(minv[v], dd);
      }
    }
  }

  // Reduce min within each 16-lane half (rows v in lanes 0-15, v+8 in 16-31).
#pragma unroll
  for (int off = 8; off >= 1; off >>= 1) {
#pragma unroll
    for (int v = 0; v < 8; ++v) minv[v] = fminf(minv[v], __shfl_xor(minv[v], off, 32));
  }
  if (m == 0) {
#pragma unroll
    for (int v = 0; v < 8; ++v) ldsmin[wave][v + hi * 8] = minv[v];
  }
  __syncthreads();
  if (tid < MROWS) {
    int slice = tid >> 4, rl = tid & 15;
    negmin_s[tid] = fminf(ldsmin[slice * 2][rl], ldsmin[slice * 2 + 1][rl]);
  }
  __syncthreads();

  float negr[8];
#pragma unroll
  for (int v = 0; v < 8; ++v) negr[v] = negmin_s[rloc0 + v + hi * 8];

  // -------------------- PASS 2: selected-pair loss sum ----------------------
  float sum = 0.0f, cnt = 0.0f;
  for (int g = colset; g < numGroups; g += 2) {
    const int colBase = g * 64;
    v8f acc[4];
#pragma unroll
    for (int t = 0; t < 4; ++t) acc[t] = vzero;
    gemm_group(acc, X, colBase, m, koff, a_base, d);
#pragma unroll
    for (int t = 0; t < 4; ++t) {
      int col   = colBase + t * 16 + m;
      float sqc = sq[col];
      int   tc  = tgt[col];
      int   cc  = cam[col];
#pragma unroll
      for (int v = 0; v < 8; ++v) {
        if (tc == trow[v]) {
          int cp = cc * crow[v];
          if (cp == 3 || cp == 6) {
            float d2 = sqrow[v] + sqc - 2.0f * acc[t][v];
            float dd = __builtin_sqrtf(fmaxf(d2, 1e-12f));
            sum += fmaxf(0.0f, MARGIN_F + dd - negr[v]);
            cnt += 1.0f;
          }
        }
      }
    }
  }

#pragma unroll
  for (int off = 16; off >= 1; off >>= 1) {
    sum += __shfl_xor(sum, off, 32);
    cnt += __shfl_xor(cnt, off, 32);
  }
  if (lane == 0) { psum[wave] = sum; pcnt[wave] = cnt; }
  __syncthreads();
  if (tid == 0) {
    float S = 0.0f, C = 0.0f;
#pragma unroll
    for (int w = 0; w < 8; ++w) { S += psum[w]; C += pcnt[w]; }
    atomicAdd(&ws[0], S);
    atomicAdd(&ws[1], C);
  }
}

// ---------------------------------------------------------------------------
__global__ void finalize_kernel(const float* __restrict__ ws, float* __restrict__ out) {
  out[0] = ws[0] / ws[1];
}

// ---------------------------------------------------------------------------
extern "C" void kernel_launch(void* const* d_in, const int* in_sizes, int n_in,
                              void* d_out, int out_size, void* d_ws, size_t ws_size,
                              hipStream_t stream) {
  const float* X   = (const float*)d_in[0];
  const int*   tgt = (const int*)d_in[1];
  const int*   cam = (const int*)d_in[2];
  float*       ws  = (float*)d_ws;
  float*       out = (float*)d_out;
  const int n = in_sizes[1];            // 8192
  const int d = in_sizes[0] / n;        // 512

  const size_t lds_bytes = (size_t)MROWS * (d + LDS_PAD) * sizeof(float);

  sq_kernel<<<(n + 7) / 8, 256, 0, stream>>>(X, ws, n, d);
  triplet_main<<<n / MROWS, 256, lds_bytes, stream>>>(X, tgt, cam, ws, n, d);
  finalize_kernel<<<1, 1, 0, stream>>>(ws, out);
}